// GraphConv_21466246545953
// MI455X (gfx1250) — compile-verified
//
#include <hip/hip_runtime.h>
#include <hip/hip_bf16.h>

// ---------------------------------------------------------------------------
// GCN layer for MI455X (gfx1250, wave32):
//   support = x @ W   -> fp32 WMMA 16x16x4, W staged through LDS in K-blocks
//   out     = seg_sum(val * support[col]) + bias -> sorted-row, atomic-free
// ---------------------------------------------------------------------------

typedef float v2f __attribute__((ext_vector_type(2)));
typedef float v8f __attribute__((ext_vector_type(8)));

#define D 256              // D_IN == D_OUT == 256
#define WAVES_PER_BLOCK 8  // 256 threads = 8 wave32
#define KB 64              // K-block staged in LDS (64 x 256 fp32 = 64 KB)

// ---------------------------------------------------------------------------
// Kernel 1: support[M x 256] = X[M x 256] @ W[256 x 256], fp32 WMMA.
//
// One wave owns a full 16x256 output strip: 16 accumulators (128 VGPRs),
// A fragment loaded ONCE per K-step and reused across all 16 N-tiles, so x is
// read exactly once from HBM. W is staged cooperatively into LDS in K-blocks
// of 64, packed as (W[k][n], W[k+1][n]) pairs so each lane's B fragment is a
// single aligned ds_load_b64.
//
// Fragment layouts (ISA 7.12.2, f32 16x16x4):
//   A: lanes 0-15 = row M=lane, {K0,K1}; lanes 16-31 = row M=lane-16, {K2,K3}
//   B: lane n (<16) = col n, {B[K0][n],B[K1][n]}; lane n+16 = {B[K2][n],B[K3][n]}
//   C/D: VGPR r = row r (lanes 0-15) / row r+8 (lanes 16-31)
// ---------------------------------------------------------------------------
__global__ void __launch_bounds__(256)
gcn_gemm_wmma_f32(const float* __restrict__ X,
                  const float* __restrict__ W,
                  float* __restrict__ S,
                  int M)
{
    // LDS slab: 32 pair-rows x 256 cols x 2 floats = 64 KB
    __shared__ float ldsW[(KB / 2) * (D * 2)];

    const int tid   = threadIdx.x;
    const int wave  = tid >> 5;
    const int lane  = tid & 31;
    const int mtile = blockIdx.x * WAVES_PER_BLOCK + wave;
    const int m0    = mtile * 16;
    const bool active = (m0 < M);              // wave-uniform

    const int half = lane >> 4;                // 0: {K0,K1}, 1: {K2,K3}
    const int l15  = lane & 15;

    v8f acc[16];
    #pragma unroll
    for (int t = 0; t < 16; ++t) acc[t] = (v8f){};

    const float* __restrict__ xrow =
        X + (size_t)((active ? m0 : 0) + l15) * D;

    for (int kb = 0; kb < D; kb += KB) {
        __syncthreads();
        // Cooperative stage: thread t writes column n=t for all 64 K rows.
        // Global reads coalesced; LDS writes stride-2 dwords -> 32 lanes map
        // onto 64 banks conflict-free.
        #pragma unroll 8
        for (int i = 0; i < KB; ++i) {
            ldsW[(i >> 1) * (D * 2) + tid * 2 + (i & 1)] =
                W[(size_t)(kb + i) * D + tid];
        }
        __syncthreads();

        if (active) {
            for (int k = 0; k < KB; k += 4) {
                // A fragment: 2 contiguous fp32 (8B aligned: k+2*half even)
                v2f a = *(const v2f*)(xrow + kb + k + 2 * half);
                const float* __restrict__ bp =
                    &ldsW[((k >> 1) + half) * (D * 2) + l15 * 2];
                #pragma unroll
                for (int t = 0; t < 16; ++t) {
                    v2f b = *(const v2f*)(bp + t * 32);   // ds_load_b64
                    acc[t] = __builtin_amdgcn_wmma_f32_16x16x4_f32(
                        false, a, false, b, (short)0, acc[t], false, false);
                }
            }
        }
    }

    if (active) {
        #pragma unroll
        for (int r = 0; r < 8; ++r) {
            float* __restrict__ srow =
                S + (size_t)(m0 + r + 8 * half) * D + l15;
            #pragma unroll
            for (int t = 0; t < 16; ++t) srow[t * 16] = acc[t][r];
        }
    }
}

// ---------------------------------------------------------------------------
// Kernel 2: per-node segment reduction (rows are SORTED -> contiguous spans).
// One 256-thread workgroup per output node; thread t owns output column t.
// Atomic-free and deterministic. (col,val) staged through LDS in 256-edge
// chunks; support (102 MB) is L2-resident (192 MB L2) so gathers run at L2
// bandwidth. global_prefetch issued for the next gathered row.
// ---------------------------------------------------------------------------
__device__ __forceinline__ long long
lbound_i64(const long long* __restrict__ a, long long n, long long key)
{
    long long lo = 0, hi = n;
    while (lo < hi) {
        long long mid = (lo + hi) >> 1;
        if (a[mid] < key) lo = mid + 1; else hi = mid;
    }
    return lo;
}

#define CHUNK 256

__global__ void __launch_bounds__(256)
gcn_spmm_rows(const float* __restrict__ support,
              const long long* __restrict__ rows,
              const long long* __restrict__ cols,
              const float* __restrict__ vals,
              const float* __restrict__ bias,
              float* __restrict__ out,
              long long nEdges)
{
    __shared__ int   s_col[CHUNK];
    __shared__ float s_val[CHUNK];

    const long long node = blockIdx.x;
    const int tid = threadIdx.x;

    // Uniform binary search: edge span [lo, hi) for this node (scalar-cached).
    const long long lo = lbound_i64(rows, nEdges, node);
    const long long hi = lbound_i64(rows, nEdges, node + 1);

    float acc = bias[tid];

    for (long long base = lo; base < hi; base += CHUNK) {
        const int cnt = (int)(((hi - base) < (long long)CHUNK) ? (hi - base)
                                                               : (long long)CHUNK);
        if (tid < cnt) {
            s_col[tid] = (int)cols[base + tid];
            s_val[tid] = vals[base + tid];
        }
        __syncthreads();

        for (int e = 0; e < cnt; ++e) {
            const int   c = s_col[e];
            const float v = s_val[e];
            if (e + 1 < cnt) {  // prefetch next gathered row (global_prefetch)
                __builtin_prefetch(&support[(size_t)s_col[e + 1] * D + tid], 0, 1);
            }
            acc = fmaf(v, support[(size_t)c * D + tid], acc);
        }
        __syncthreads();
    }

    out[node * D + tid] = acc;
}

// ---------------------------------------------------------------------------
// Launch: inputs per reference order:
//   0: x [N*256] f32   1: adj_row [E] i64   2: adj_col [E] i64
//   3: adj_val [E] f32 4: weight [256*256] f32   5: bias [256] f32
// d_ws holds support (N*256 f32 = 102.4 MB).
// ---------------------------------------------------------------------------
extern "C" void kernel_launch(void* const* d_in, const int* in_sizes, int n_in,
                              void* d_out, int out_size, void* d_ws, size_t ws_size,
                              hipStream_t stream)
{
    const float*     x      = (const float*)d_in[0];
    const long long* arow   = (const long long*)d_in[1];
    const long long* acol   = (const long long*)d_in[2];
    const float*     aval   = (const float*)d_in[3];
    const float*     weight = (const float*)d_in[4];
    const float*     bias   = (const float*)d_in[5];
    float*           out    = (float*)d_out;
    float*           support= (float*)d_ws;

    const long long N = in_sizes[0] / D;     // 100000 (multiple of 16)
    const long long E = in_sizes[3];         // 3200000

    // GEMM: support = x @ W. One wave per 16-row strip (full 256-wide output).
    const int mtiles     = (int)(N / 16);                        // 6250
    const int gemmBlocks = (mtiles + WAVES_PER_BLOCK - 1) / WAVES_PER_BLOCK;
    hipLaunchKernelGGL(gcn_gemm_wmma_f32, dim3(gemmBlocks), dim3(256), 0, stream,
                       x, weight, support, (int)N);

    // SpMM + bias: one workgroup per node, deterministic segment sum.
    hipLaunchKernelGGL(gcn_spmm_rows, dim3((unsigned)N), dim3(256), 0, stream,
                       support, arow, acol, aval, bias, out, E);
}